// SSIMLoss_77764677861729
// MI455X (gfx1250) — compile-verified
//
#include <hip/hip_runtime.h>

// ---------------------------------------------------------------------------
// SSIM loss on MI455X (gfx1250). 11x11 box sums are expressed as banded 0/1
// matrix multiplies on V_WMMA_F32_16X16X4_F32 (fp32 WMMA, full precision).
// Round 2: product planes precomputed in LDS (no switch/guards in the WMMA
// loops), band fragments hoisted to registers, conflict-free LDS strides.
// ---------------------------------------------------------------------------

typedef __attribute__((ext_vector_type(2))) float v2f;
typedef __attribute__((ext_vector_type(8))) float v8f;

#define IMG      512
#define TILE_H   32
#define TILE_W   16
#define HALO_H   42          // TILE_H + 10 (valid halo rows)
#define HALO_W   26          // TILE_W + 10 (valid halo cols)
#define Q_ROWS   48          // halo rows padded to 3 row-blocks of 16
#define Q_COLS   28          // halo cols padded to K reach (4*6+2+1 = 27)
#define HPADW    18          // col stride for H/V arrays (bank-conflict free)
#define SSIM_C1  1.0e-4f
#define SSIM_C2  9.0e-4f
#define TILES_PER_PLANE 512  // (512/32)*(512/16)
#define NT       256         // threads per block (8 waves)

__global__ __launch_bounds__(NT) void ssim_tile_kernel(
    const float* __restrict__ pred, const float* __restrict__ targ,
    float* __restrict__ tsums)
{
    __shared__ float sQ[5][Q_ROWS][Q_COLS];  // p, t, p*p, t*t, p*t (zero-padded)
    __shared__ float sH[5][Q_ROWS][HPADW];   // horizontal box sums
    __shared__ float sV[5][TILE_H][HPADW];   // vertical box sums / 121
    __shared__ float sRed[NT];

    const int tid  = threadIdx.x;
    const int lane = tid & 31;
    const int wave = tid >> 5;
    const int mn   = lane & 15;   // M (A rows / D rows) or N (B cols / D cols)
    const int half = lane >> 4;   // lane group: selects K pair / row half

    const int plane = blockIdx.z;                // n*3 + c
    const int row0  = (int)blockIdx.y * TILE_H - 5;
    const int col0  = (int)blockIdx.x * TILE_W - 5;
    const float* pbase = pred + (size_t)plane * IMG * IMG;
    const float* tbase = targ + (size_t)plane * IMG * IMG;

    // ---- stage halo products into LDS, zero-padded over the full A footprint
    for (int i = tid; i < Q_ROWS * Q_COLS; i += NT) {
        const int hr = i / Q_COLS, hc = i % Q_COLS;
        const int gr = row0 + hr, gc = col0 + hc;
        float pv = 0.f, tv = 0.f;
        if (hr < HALO_H && hc < HALO_W &&
            gr >= 0 && gr < IMG && gc >= 0 && gc < IMG) {
            const size_t off = (size_t)gr * IMG + gc;
            pv = pbase[off];
            tv = tbase[off];
        }
        sQ[0][hr][hc] = pv;
        sQ[1][hr][hc] = tv;
        sQ[2][hr][hc] = pv * pv;
        sQ[3][hr][hc] = tv * tv;
        sQ[4][hr][hc] = pv * tv;
    }

    // ---- band fragments: identical for horizontal-B and vertical-A --------
    // band[s] element (K, x) = 1 iff 0 <= K - x <= 10, K = 4s + 2*half (+1)
    v2f band[7];
    #pragma unroll
    for (int s = 0; s < 7; ++s) {
        const int d0 = 4 * s + 2 * half - mn;
        band[s].x = (d0 >= 0 && d0 <= 10) ? 1.f : 0.f;
        band[s].y = (d0 + 1 >= 0 && d0 + 1 <= 10) ? 1.f : 0.f;
    }

    __syncthreads();

    // ---- horizontal pass: H[q](48x16) = Q[q](48x28) * Band(28x16) ---------
    // 15 units = 5 quantities x 3 row-blocks; 7 K=4 WMMA slices each.
    for (int u = wave; u < 15; u += 8) {
        const int q  = u / 3;
        const int rb = u - 3 * q;
        const float* arow = &sQ[q][16 * rb + mn][0];
        v8f acc = {};
        #pragma unroll
        for (int s = 0; s < 7; ++s) {
            const int ac = 4 * s + 2 * half;   // even -> 8B-aligned pair
            v2f a;
            a.x = arow[ac];
            a.y = arow[ac + 1];
            acc = __builtin_amdgcn_wmma_f32_16x16x4_f32(
                false, a, false, band[s], (short)0, acc, false, false);
        }
        #pragma unroll
        for (int v = 0; v < 8; ++v)
            sH[q][16 * rb + v + 8 * half][mn] = acc[v];
    }
    __syncthreads();

    // ---- vertical pass: V[q](32x16) = Band(16x28) * H[q](28x16) -----------
    // 10 units = 5 quantities x 2 row-blocks; 7 K=4 WMMA slices each.
    for (int u = wave; u < 10; u += 8) {
        const int q  = u >> 1;
        const int rb = u & 1;
        v8f acc = {};
        #pragma unroll
        for (int s = 0; s < 7; ++s) {
            const int hr = 16 * rb + 4 * s + 2 * half;  // <= 43 < 48
            v2f b;
            b.x = sH[q][hr][mn];
            b.y = sH[q][hr + 1][mn];
            acc = __builtin_amdgcn_wmma_f32_16x16x4_f32(
                false, band[s], false, b, (short)0, acc, false, false);
        }
        #pragma unroll
        for (int v = 0; v < 8; ++v)
            sV[q][16 * rb + v + 8 * half][mn] = acc[v] * (1.f / 121.f);
    }
    __syncthreads();

    // ---- pointwise SSIM + tile reduction -----------------------------------
    float lsum = 0.f;
    for (int i = tid; i < TILE_H * TILE_W; i += NT) {
        const int r = i >> 4, c = i & 15;
        const float mux = sV[0][r][c], muy = sV[1][r][c];
        const float pxx = sV[2][r][c], pyy = sV[3][r][c], pxy = sV[4][r][c];
        const float mux2 = mux * mux, muy2 = muy * muy, muxy = mux * muy;
        const float sx  = pxx - mux2, sy = pyy - muy2, sxy = pxy - muxy;
        const float num = (2.f * muxy + SSIM_C1) * (2.f * sxy + SSIM_C2);
        const float den = (mux2 + muy2 + SSIM_C1) * (sx + sy + SSIM_C2);
        float ssim = num / den;
        ssim = fminf(fmaxf(ssim, 0.f), 1.f);
        lsum += ssim;
    }
    sRed[tid] = lsum;
    __syncthreads();
    #pragma unroll
    for (int off = NT / 2; off > 0; off >>= 1) {
        if (tid < off) sRed[tid] += sRed[tid + off];
        __syncthreads();
    }
    if (tid == 0) {
        const int tileIdx = (int)blockIdx.y * (int)gridDim.x + (int)blockIdx.x;
        tsums[(size_t)plane * TILES_PER_PLANE + tileIdx] = sRed[0];
    }
}

// Deterministic second-stage reduction: one block per sample sums its
// 3 * 512 tile sums in a fixed order -> bit-stable across graph replays.
__global__ __launch_bounds__(256) void ssim_reduce_kernel(
    const float* __restrict__ tsums, float* __restrict__ out)
{
    __shared__ float red[256];
    const int n = blockIdx.x, t = threadIdx.x;
    const float* base = tsums + (size_t)n * 3 * TILES_PER_PLANE;
    float s = 0.f;
    for (int i = t; i < 3 * TILES_PER_PLANE; i += 256) s += base[i];
    red[t] = s;
    __syncthreads();
    #pragma unroll
    for (int off = 128; off > 0; off >>= 1) {
        if (t < off) red[t] += red[t + off];
        __syncthreads();
    }
    if (t == 0)
        out[n] = 1.f - red[0] * (1.f / (3.f * IMG * IMG));
}

extern "C" void kernel_launch(void* const* d_in, const int* in_sizes, int n_in,
                              void* d_out, int out_size, void* d_ws, size_t ws_size,
                              hipStream_t stream) {
    const float* pred = (const float*)d_in[0];
    const float* targ = (const float*)d_in[1];
    float* out   = (float*)d_out;
    float* tsums = (float*)d_ws;   // 48 planes * 512 tiles * 4B = 98304 B

    dim3 grid(IMG / TILE_W, IMG / TILE_H, 16 * 3);  // (32, 16, 48)
    ssim_tile_kernel<<<grid, NT, 0, stream>>>(pred, targ, tsums);
    ssim_reduce_kernel<<<16, 256, 0, stream>>>(tsums, out);
}